// CrossAdjacencyMatrix_56392920596510
// MI455X (gfx1250) — compile-verified
//
#include <hip/hip_runtime.h>
#include <hip/hip_bf16.h>

// Problem constants (match reference).
#define E_DIM  5000
#define N_TRI  200000
#define R_DIM  1000
#define D_DIM  128
// 1 / (3 * sqrt(128))
#define TRANSE_INV 0.02946278254943948f

typedef __attribute__((ext_vector_type(2))) float v2f;
typedef __attribute__((ext_vector_type(8))) float v8f;

// ---------------------------------------------------------------------------
// Kernel 1: zero-fill the two E x E output matrices (float4 streaming stores).
// 2*E*E = 50,000,000 floats = 12,500,000 float4 (divisible by 4).
// ---------------------------------------------------------------------------
__global__ void zero_out_kernel(float4* __restrict__ out, long n4) {
    long i      = (long)blockIdx.x * blockDim.x + threadIdx.x;
    long stride = (long)gridDim.x * blockDim.x;
    float4 z = make_float4(0.f, 0.f, 0.f, 0.f);
    for (; i < n4; i += stride) out[i] = z;
}

// ---------------------------------------------------------------------------
// Kernel 2: per-triple sparse update.
// Each wave handles 16 triples. The D=128 L1-distance reduction is done on the
// matrix pipe: v_wmma_f32_16x16x4_f32 with an all-ones B matrix accumulates
// row sums of A = |he + re - te| (exact f32 math). 32 WMMA steps cover K=128.
//
// A-fragment (16x4 f32): lanes 0-15 hold K=kb+0,kb+1; lanes 16-31 hold
// K=kb+2,kb+3, with M = lane & 15. D-fragment: lanes 0-15 VGPR v holds row
// M=v, lanes 16-31 VGPR v holds row M=8+v (all 16 columns identical since B
// is all ones). So lane 0 owns the 8 sums for triples t0+0..7 and lane 16
// owns triples t0+8..15, each in statically indexed registers.
// ---------------------------------------------------------------------------
__global__ __launch_bounds__(256) void triple_update_kernel(
    const float* __restrict__ ent,   // [E, D]
    const float* __restrict__ rel,   // [R, D]
    const float* __restrict__ relw,  // [R]
    const float* __restrict__ conf,  // [E, E]
    const float* __restrict__ imp,   // [E, E]
    const float* __restrict__ pca,   // [E, E]
    const int*   __restrict__ head,  // [N]
    const int*   __restrict__ tail,  // [N]
    const int*   __restrict__ rvec,  // [N]
    float*       __restrict__ out)   // [E, E]
{
    const int lane = threadIdx.x & 31;
    const int wave = (int)((blockIdx.x * (long)blockDim.x + threadIdx.x) >> 5);
    const int t0   = wave * 16;
    if (t0 >= N_TRI) return;               // wave-uniform (EXEC stays all-1s)

    const int m     = lane & 15;           // triple within group
    const int khalf = lane >> 4;           // 0: K=kb+0..1, 1: K=kb+2..3
    const int trip  = t0 + m;              // always < N (N % 16 == 0)

    const int h = head[trip];
    const int t = tail[trip];
    const int r = rvec[trip];

    const float2* he2 = (const float2*)(ent + (size_t)h * D_DIM);
    const float2* te2 = (const float2*)(ent + (size_t)t * D_DIM);
    const float2* re2 = (const float2*)(rel + (size_t)r * D_DIM);

    v2f ones; ones.x = 1.0f; ones.y = 1.0f;      // B matrix = all ones
    v8f acc = {0.f, 0.f, 0.f, 0.f, 0.f, 0.f, 0.f, 0.f};

    #pragma unroll 8
    for (int kb = 0; kb < D_DIM; kb += 4) {
        const int kk = (kb >> 1) + khalf;        // float2 index
        const float2 a0 = he2[kk];
        const float2 b0 = re2[kk];
        const float2 c0 = te2[kk];
        v2f a;
        a.x = fabsf(a0.x + b0.x - c0.x);
        a.y = fabsf(a0.y + b0.y - c0.y);
        acc = __builtin_amdgcn_wmma_f32_16x16x4_f32(
                  false, a, false, ones, (short)0, acc, false, false);
    }

    // Lanes 0 and 16 each own 8 complete row sums; they perform the sparse
    // scatter stores. (h,t) positions are unique per graph, so plain stores
    // into the zero-filled output are race-free.
    if (lane == 0 || lane == 16) {
        const int mbase = khalf * 8;
        #pragma unroll
        for (int v = 0; v < 8; ++v) {
            const int ti = t0 + mbase + v;
            const int hh = head[ti];
            const int tt = tail[ti];
            const int rr = rvec[ti];
            const size_t p = (size_t)hh * E_DIM + tt;
            const float tv = 1.0f - acc[v] * TRANSE_INV;
            out[p] = conf[p] * imp[p] *
                     (0.3f * pca[p] + 0.3f * tv + 0.4f * relw[rr]);
        }
    }
}

// ---------------------------------------------------------------------------
// Kernel 3: add sparse unit matrix (runs after triple kernel; stream-ordered).
// ---------------------------------------------------------------------------
__global__ void diag_add_kernel(float* __restrict__ out) {
    int i = blockIdx.x * blockDim.x + threadIdx.x;
    if (i < E_DIM) {
        size_t p = (size_t)i * E_DIM + i;
        out[p] += 1.0f;
    }
}

extern "C" void kernel_launch(void* const* d_in, const int* in_sizes, int n_in,
                              void* d_out, int out_size, void* d_ws, size_t ws_size,
                              hipStream_t stream) {
    (void)in_sizes; (void)n_in; (void)out_size; (void)d_ws; (void)ws_size;

    const float* ent_sr  = (const float*)d_in[0];
    const float* ent_tg  = (const float*)d_in[1];
    const float* rel_sr  = (const float*)d_in[2];
    const float* rel_tg  = (const float*)d_in[3];
    const float* relw_sr = (const float*)d_in[4];
    const float* relw_tg = (const float*)d_in[5];
    const float* conf_sr = (const float*)d_in[6];
    const float* imp_sr  = (const float*)d_in[7];
    const float* pca_sr  = (const float*)d_in[8];
    const float* conf_tg = (const float*)d_in[9];
    const float* imp_tg  = (const float*)d_in[10];
    const float* pca_tg  = (const float*)d_in[11];
    const int*   head_sr = (const int*)d_in[12];
    const int*   tail_sr = (const int*)d_in[13];
    const int*   r_sr    = (const int*)d_in[14];
    const int*   head_tg = (const int*)d_in[15];
    const int*   tail_tg = (const int*)d_in[16];
    const int*   r_tg    = (const int*)d_in[17];

    float* out_sr = (float*)d_out;
    float* out_tg = out_sr + (size_t)E_DIM * E_DIM;

    // 1) zero-fill both output matrices (bulk of HBM traffic: ~200 MB stores)
    const long n4 = (2L * E_DIM * E_DIM) / 4;
    zero_out_kernel<<<4096, 256, 0, stream>>>((float4*)d_out, n4);

    // 2) sparse triple updates (16 triples per wave)
    const int waves  = N_TRI / 16;               // 12500
    const int blocks = (waves * 32 + 255) / 256; // 1563
    triple_update_kernel<<<blocks, 256, 0, stream>>>(
        ent_sr, rel_sr, relw_sr, conf_sr, imp_sr, pca_sr,
        head_sr, tail_sr, r_sr, out_sr);
    triple_update_kernel<<<blocks, 256, 0, stream>>>(
        ent_tg, rel_tg, relw_tg, conf_tg, imp_tg, pca_tg,
        head_tg, tail_tg, r_tg, out_tg);

    // 3) + sparse unit matrix
    diag_add_kernel<<<(E_DIM + 255) / 256, 256, 0, stream>>>(out_sr);
    diag_add_kernel<<<(E_DIM + 255) / 256, 256, 0, stream>>>(out_tg);
}